// PointPillarsScatter_82660940579133
// MI455X (gfx1250) — compile-verified
//
#include <hip/hip_runtime.h>

// PointPillarsScatter for MI455X (gfx1250, wave32).
// Memory-bound scatter: no matmul -> no WMMA (nothing to feed matrix units).
// Optimized for the L2-resident path: the entire 81.4 MB canvas fits in the
// 192 MB L2, so regular-temporal 128-bit stores keep it resident; scatter
// writes then hit L2. Duplicates resolved deterministically via
// global_atomic_max_i32 ("largest pillar index wins" == sequential
// last-write-wins).

#define B_ 4
#define C_ 64
#define H_ 282
#define W_ 282
#define P_ 12000

static constexpr float kPillar   = 0.16f;
static constexpr float kRangeMin = -22.0f;

// ---------------------------------------------------------------------------
// Phase 0: zero the output canvas + init owner map. One-shot, 4x-unrolled:
// each thread issues up to 4 predicated global_store_b128, block-interleaved
// so every store instruction covers a contiguous 4 KB span per 256-thread
// block. No grid-stride loop -> minimal SALU overhead per byte in the phase
// that dominates the roofline.
// ---------------------------------------------------------------------------
__global__ void pp_init_kernel(float4* __restrict__ out, int n4_out,
                               int4* __restrict__ owner, int n4_own) {
    const int bdim = blockDim.x;                       // 256
    const int base = blockIdx.x * (bdim * 4) + threadIdx.x;
    const float4 z4 = make_float4(0.0f, 0.0f, 0.0f, 0.0f);

    const int i0 = base;
    const int i1 = base + bdim;
    const int i2 = base + 2 * bdim;
    const int i3 = base + 3 * bdim;
    if (i0 < n4_out) out[i0] = z4;
    if (i1 < n4_out) out[i1] = z4;
    if (i2 < n4_out) out[i2] = z4;
    if (i3 < n4_out) out[i3] = z4;

    // Owner map: 79,524 int4 quads < total thread count -> single store.
    const int tid = blockIdx.x * bdim + threadIdx.x;
    if (tid < n4_own) owner[tid] = make_int4(-1, -1, -1, -1);
}

// ---------------------------------------------------------------------------
// Per-pillar grid-cell computation, matching the JAX reference exactly:
//   valid = (x != 0); xg = clip(floor((x-RMIN)/SZ), 0, W-1) when valid,
//   invalid pillars map out of bounds and are dropped by the scatter.
// Returns cell offset (y*W + x) or -1 if invalid.
// ---------------------------------------------------------------------------
__device__ __forceinline__ int pp_cell(const float* __restrict__ pfn_in,
                                       int b, int p) {
    const float x = pfn_in[(size_t)b * (2 * P_) + p];         // [b,0,p,0]
    const float y = pfn_in[(size_t)b * (2 * P_) + P_ + p];    // [b,1,p,0]
    if (x == 0.0f) return -1;
    int xg = (int)floorf((x - kRangeMin) / kPillar);
    int yg = (int)floorf((y - kRangeMin) / kPillar);
    xg = min(max(xg, 0), W_ - 1);
    yg = min(max(yg, 0), H_ - 1);
    return yg * W_ + xg;
}

// ---------------------------------------------------------------------------
// Phase 1: each valid pillar claims its cell with atomicMax(pillar index).
// Emits global_atomic_max_i32 without return (STOREcnt path).
// ---------------------------------------------------------------------------
__global__ void pp_claim_kernel(const float* __restrict__ pfn_in,
                                int* __restrict__ owner) {
    const int t = blockIdx.x * blockDim.x + threadIdx.x;
    if (t >= B_ * P_) return;
    const int b = t / P_;
    const int p = t - b * P_;
    const int cell = pp_cell(pfn_in, b, p);
    if (cell < 0) return;
    atomicMax(&owner[b * (H_ * W_) + cell], p);
}

// ---------------------------------------------------------------------------
// Phase 2: resolve each pillar's destination: cell offset if it won, else -1.
// ---------------------------------------------------------------------------
__global__ void pp_resolve_kernel(const float* __restrict__ pfn_in,
                                  const int* __restrict__ owner,
                                  int* __restrict__ dest) {
    const int t = blockIdx.x * blockDim.x + threadIdx.x;
    if (t >= B_ * P_) return;
    const int b = t / P_;
    const int p = t - b * P_;
    const int cell = pp_cell(pfn_in, b, p);
    int d = -1;
    if (cell >= 0 && owner[b * (H_ * W_) + cell] == p) d = cell;
    dest[t] = d;
}

// ---------------------------------------------------------------------------
// Phase 3: scatter winning feature columns.
// Thread t handles 4 consecutive pillars of one (b,c): feature and dest
// loads are 128-bit and fully coalesced (global_load_b128); the 4B canvas
// stores scatter but land in L2 (canvas is resident from phase 0).
// ---------------------------------------------------------------------------
__global__ void pp_scatter_kernel(const float4* __restrict__ feats4,   // (B*C, P/4)
                                  const int4*  __restrict__ dest4,     // (B,   P/4)
                                  float* __restrict__ out) {           // (B*C, H*W)
    const int PQ = P_ / 4;
    const int t = blockIdx.x * blockDim.x + threadIdx.x;
    if (t >= B_ * C_ * PQ) return;

    const int q  = t % PQ;          // pillar quad
    const int bc = t / PQ;          // fused batch*channel
    const int b  = bc / C_;

    const int4   d = dest4[b * PQ + q];   // coalesced 16B load
    const float4 f = feats4[t];           // coalesced 16B load

    float* __restrict__ plane = out + (size_t)bc * (H_ * W_);
    if (d.x >= 0) plane[d.x] = f.x;
    if (d.y >= 0) plane[d.y] = f.y;
    if (d.z >= 0) plane[d.z] = f.z;
    if (d.w >= 0) plane[d.w] = f.w;
}

// ---------------------------------------------------------------------------
// Launcher
// ---------------------------------------------------------------------------
extern "C" void kernel_launch(void* const* d_in, const int* in_sizes, int n_in,
                              void* d_out, int out_size, void* d_ws, size_t ws_size,
                              hipStream_t stream) {
    (void)in_sizes; (void)n_in; (void)ws_size;

    const float* pfn_in  = (const float*)d_in[0];   // (B, 2, P, 1) f32
    const float* pfn_out = (const float*)d_in[1];   // (B, C, P)    f32
    float*       out     = (float*)d_out;           // (B, C, H, W) f32

    // Workspace layout: [ owner: B*H*W ints | dest: B*P ints ]  (~1.46 MB)
    int* owner = (int*)d_ws;
    int* dest  = owner + (size_t)B_ * H_ * W_;

    const int n4_out = out_size / 4;                 // 5,089,536 float4s
    const int n4_own = (B_ * H_ * W_) / 4;           // 79,524 int4s

    // Phase 0: zero canvas + owner init, one-shot 4x-unrolled b128 stores.
    {
        const int threads = 256;                     // 8 wave32 waves/block
        const int blocks  = (n4_out + threads * 4 - 1) / (threads * 4);  // 4971
        pp_init_kernel<<<blocks, threads, 0, stream>>>(
            (float4*)out, n4_out, (int4*)owner, n4_own);
    }

    // Phase 1: claim cells (B*P = 48,000 pillars).
    {
        const int threads = 256;
        const int blocks  = (B_ * P_ + threads - 1) / threads;
        pp_claim_kernel<<<blocks, threads, 0, stream>>>(pfn_in, owner);
    }

    // Phase 2: resolve winners into per-pillar destinations.
    {
        const int threads = 256;
        const int blocks  = (B_ * P_ + threads - 1) / threads;
        pp_resolve_kernel<<<blocks, threads, 0, stream>>>(pfn_in, owner, dest);
    }

    // Phase 3: vectorized scatter (B*C*P/4 = 768,000 threads).
    {
        const int total   = B_ * C_ * (P_ / 4);
        const int threads = 256;
        const int blocks  = (total + threads - 1) / threads;
        pp_scatter_kernel<<<blocks, threads, 0, stream>>>(
            (const float4*)pfn_out, (const int4*)dest, out);
    }
}